// GRBlockNorm_76879914598740
// MI455X (gfx1250) — compile-verified
//
#include <hip/hip_runtime.h>
#include <math.h>

#define BLK 256
#define NEG_SLOPE 0.2f
#define LN_EPS 1e-5f

typedef float v2f __attribute__((ext_vector_type(2)));
typedef float v8f __attribute__((ext_vector_type(8)));

// ---------------------------------------------------------------- fill
__global__ void k_fill(float* __restrict__ p, long n, float v) {
  long i = (long)blockIdx.x * blockDim.x + threadIdx.x;
  long stride = (long)gridDim.x * blockDim.x;
  for (; i < n; i += stride) p[i] = v;
}

// ---------------------------------------------------------------- GEMM
// C[nrows,128] = A[nrows,128] @ W[128,128] in fp32 via V_WMMA_F32_16X16X4_F32.
// 8 waves / block, each wave owns one 16x16 output tile (block = 16 rows x 128 cols).
// Layouts (ISA 7.12.2): A 16x4: lane=M (0..15 per half), vgpr=K%2, lane-half adds 2 to K.
//                       C 16x16: vgpr r -> M=r (lanes 0-15), M=r+8 (lanes 16-31), N=lane%16.
__global__ void k_gemm128_wmma(const float* __restrict__ A,
                               const float* __restrict__ W,
                               float* __restrict__ C, int nrows) {
  const int wave = threadIdx.x >> 5;   // 0..7  -> column tile
  const int lane = threadIdx.x & 31;
  const int m    = lane & 15;
  const int hi   = lane >> 4;          // lane half
  const int row0 = blockIdx.x * 16;
  const int col0 = wave * 16;

  int ar = row0 + m; if (ar >= nrows) ar = nrows - 1;   // clamp (keeps EXEC full)
  const float* arow = A + (long)ar * 128;

  v8f acc = {};
#pragma unroll
  for (int k0 = 0; k0 < 128; k0 += 4) {
    const int ka = k0 + 2 * hi;
    v2f a; a.x = arow[ka];                 a.y = arow[ka + 1];
    v2f b; b.x = W[ka * 128 + col0 + m];   b.y = W[(ka + 1) * 128 + col0 + m];
    acc = __builtin_amdgcn_wmma_f32_16x16x4_f32(false, a, false, b,
                                                (short)0, acc, false, false);
  }
#pragma unroll
  for (int r = 0; r < 8; ++r) {
    const int row = row0 + r + 8 * hi;
    if (row < nrows) C[(long)row * 128 + col0 + m] = acc[r];
  }
}

// ---------------------------------------------------------------- attention scores
template <int H, int C>
__global__ void k_att(const float* __restrict__ h,
                      const float* __restrict__ att_src,
                      const float* __restrict__ att_dst,
                      float* __restrict__ asrc, float* __restrict__ adst, int n) {
  int t = blockIdx.x * blockDim.x + threadIdx.x;
  if (t >= n * H) return;
  const int node = t / H, hd = t % H;
  const float* row = h + (long)node * (H * C) + hd * C;
  float s1 = 0.f, s2 = 0.f;
#pragma unroll 8
  for (int c = 0; c < C; ++c) {
    const float v = row[c];
    s1 += v * att_src[hd * C + c];
    s2 += v * att_dst[hd * C + c];
  }
  asrc[t] = s1; adst[t] = s2;
}

__device__ __forceinline__ void edge_sd(int e, int eorig, const int* __restrict__ src,
                                        const int* __restrict__ dst, int& s, int& d) {
  if (e < eorig) { s = src[e]; d = dst[e]; }
  else           { s = e - eorig; d = s; }          // implicit self-loops
}

// ---------------------------------------------------------------- edge max
template <int H>
__global__ void k_edge_max(const int* __restrict__ src, const int* __restrict__ dst,
                           const float* __restrict__ asrc, const float* __restrict__ adst,
                           float* __restrict__ emax, int eorig, int etot) {
  int e = blockIdx.x * blockDim.x + threadIdx.x;
  if (e >= etot) return;
  int s, d; edge_sd(e, eorig, src, dst, s, d);
#pragma unroll
  for (int h = 0; h < H; ++h) {
    float v = asrc[s * H + h] + adst[d * H + h];
    v = v > 0.f ? v : NEG_SLOPE * v;
    atomicMax(&emax[d * H + h], v);
  }
}

// ---------------------------------------------------------------- edge exp + denom
template <int H>
__global__ void k_edge_exp(const int* __restrict__ src, const int* __restrict__ dst,
                           const float* __restrict__ asrc, const float* __restrict__ adst,
                           const float* __restrict__ emax,
                           float* __restrict__ exb, float* __restrict__ denom,
                           int eorig, int etot) {
  int e = blockIdx.x * blockDim.x + threadIdx.x;
  if (e >= etot) return;
  int s, d; edge_sd(e, eorig, src, dst, s, d);
#pragma unroll
  for (int h = 0; h < H; ++h) {
    float v = asrc[s * H + h] + adst[d * H + h];
    v = v > 0.f ? v : NEG_SLOPE * v;
    const float ex = __expf(v - emax[d * H + h]);
    exb[(long)e * H + h] = ex;
    atomicAdd(&denom[d * H + h], ex);
  }
}

// ---------------------------------------------------------------- edge scatter-aggregate
// 4 threads per edge, each handles 32 contiguous feature dims of the 128-wide row.
template <int H, int C>
__global__ void k_edge_aggr(const int* __restrict__ src, const int* __restrict__ dst,
                            const float* __restrict__ hmat,
                            const float* __restrict__ exb, const float* __restrict__ denom,
                            float* __restrict__ out, int eorig, int etot) {
  int t = blockIdx.x * blockDim.x + threadIdx.x;
  if (t >= etot * 4) return;
  const int e = t >> 2, q = t & 3;
  int s, d; edge_sd(e, eorig, src, dst, s, d);
  const int h = (q * 32) / C;
  const float alpha = exb[(long)e * H + h] / (denom[d * H + h] + 1e-16f);
  const float* hs = hmat + (long)s * 128 + q * 32;
  float* od = out + (long)d * 128 + q * 32;
  __builtin_prefetch(hs, 0, 0);
#pragma unroll
  for (int i = 0; i < 32; ++i) atomicAdd(&od[i], hs[i] * alpha);
}

// ---------------------------------------------------------------- bias add + global stats
__global__ void k_bias_stats(float* __restrict__ buf, const float* __restrict__ bias,
                             int n, float* __restrict__ stats) {
  __shared__ float ss[BLK], sq[BLK];
  const int i = blockIdx.x * blockDim.x + threadIdx.x;
  float v = 0.f;
  if (i < n) { v = buf[i] + bias[i & 127]; buf[i] = v; }
  ss[threadIdx.x] = v; sq[threadIdx.x] = v * v;
  __syncthreads();
  for (int s = BLK / 2; s > 0; s >>= 1) {
    if ((int)threadIdx.x < s) { ss[threadIdx.x] += ss[threadIdx.x + s];
                                sq[threadIdx.x] += sq[threadIdx.x + s]; }
    __syncthreads();
  }
  if (threadIdx.x == 0) { atomicAdd(&stats[0], ss[0]); atomicAdd(&stats[1], sq[0]); }
}

// ---------------------------------------------------------------- graph layernorm apply
template <bool RELU>
__global__ void k_ln_apply(float* __restrict__ buf, const float* __restrict__ w,
                           const float* __restrict__ b, const float* __restrict__ stats,
                           int n) {
  const int i = blockIdx.x * blockDim.x + threadIdx.x;
  if (i >= n) return;
  const float inv_n = 1.f / (float)n;
  const float mean = stats[0] * inv_n;
  const float var  = stats[1] * inv_n - mean * mean;
  const float rs   = rsqrtf(var + LN_EPS);
  float v = (buf[i] - mean) * rs * w[i & 127] + b[i & 127];
  if (RELU) v = fmaxf(v, 0.f);
  buf[i] = v;
}

// ---------------------------------------------------------------- SimpleConv mean + residual
__global__ void k_cnt(const int* __restrict__ dst, float* __restrict__ cnt, int e) {
  int t = blockIdx.x * blockDim.x + threadIdx.x;
  if (t >= e) return;
  atomicAdd(&cnt[dst[t]], 1.f);
}

__global__ void k_sc_aggr(const int* __restrict__ src, const int* __restrict__ dst,
                          const float* __restrict__ h, float* __restrict__ out, int e) {
  int t = blockIdx.x * blockDim.x + threadIdx.x;
  if (t >= e * 4) return;
  const int ed = t >> 2, q = t & 3;
  const int s = src[ed], d = dst[ed];
  const float* hs = h + (long)s * 128 + q * 32;
  float* od = out + (long)d * 128 + q * 32;
  __builtin_prefetch(hs, 0, 0);
#pragma unroll
  for (int i = 0; i < 32; ++i) atomicAdd(&od[i], hs[i]);
}

__global__ void k_final(float* __restrict__ out, const float* __restrict__ cnt,
                        const float* __restrict__ x, int n) {
  const int i = blockIdx.x * blockDim.x + threadIdx.x;
  if (i >= n) return;
  const int node = i >> 7;
  const float c = fmaxf(cnt[node], 1.f);
  out[i] = fmaxf(out[i] / c + x[i], 0.f);
}

// ---------------------------------------------------------------- launcher
extern "C" void kernel_launch(void* const* d_in, const int* in_sizes, int n_in,
                              void* d_out, int out_size, void* d_ws, size_t ws_size,
                              hipStream_t stream) {
  const float* x    = (const float*)d_in[0];
  const int*   eidx = (const int*)d_in[1];
  const float* W1   = (const float*)d_in[2];
  const float* as1  = (const float*)d_in[3];
  const float* ad1  = (const float*)d_in[4];
  const float* b1   = (const float*)d_in[5];
  const float* ln1w = (const float*)d_in[6];
  const float* ln1b = (const float*)d_in[7];
  const float* W2   = (const float*)d_in[8];
  const float* as2  = (const float*)d_in[9];
  const float* ad2  = (const float*)d_in[10];
  const float* b2   = (const float*)d_in[11];
  const float* ln2w = (const float*)d_in[12];
  const float* ln2b = (const float*)d_in[13];
  float* out = (float*)d_out;

  const int n    = in_sizes[0] / 128;   // 50000
  const int e    = in_sizes[1] / 2;     // 800000
  const int etot = e + n;               // + self-loops
  const int* esrc = eidx;
  const int* edst = eidx + e;

  float* ws   = (float*)d_ws;
  float* hbuf = ws;  ws += (long)n * 128;   // GEMM output (h1, later h2)
  float* o1   = ws;  ws += (long)n * 128;   // layer-1 aggregated / normalized
  float* o2   = ws;  ws += (long)n * 128;   // layer-2 aggregated / normalized
  float* asb  = ws;  ws += (long)n * 2;
  float* adb  = ws;  ws += (long)n * 2;
  float* emx  = ws;  ws += (long)n * 2;
  float* den  = ws;  ws += (long)n * 2;
  float* exb  = ws;  ws += (long)etot * 2;
  float* cntb = ws;  ws += n;
  float* stats= ws;  ws += 2;

  dim3 blk(BLK);
  const int gElem = (n * 128 + BLK - 1) / BLK;
  const int gTile = (n + 15) / 16;
  const int gEdge = (etot + BLK - 1) / BLK;
  const int gAggr = (etot * 4 + BLK - 1) / BLK;
  const int gN2   = (n * 2 + BLK - 1) / BLK;
  const int gN1   = (n + BLK - 1) / BLK;
  const int gE    = (e + BLK - 1) / BLK;
  const int gE4   = (e * 4 + BLK - 1) / BLK;

  // ---------------- layer 1: GATConv(128 -> 64, heads=2, concat) ----------------
  k_gemm128_wmma<<<gTile, blk, 0, stream>>>(x, W1, hbuf, n);
  k_att<2, 64><<<gN2, blk, 0, stream>>>(hbuf, as1, ad1, asb, adb, n);
  k_fill<<<512, blk, 0, stream>>>(emx, (long)n * 2, -1e30f);
  k_fill<<<512, blk, 0, stream>>>(den, (long)n * 2, 0.f);
  k_edge_max<2><<<gEdge, blk, 0, stream>>>(esrc, edst, asb, adb, emx, e, etot);
  k_edge_exp<2><<<gEdge, blk, 0, stream>>>(esrc, edst, asb, adb, emx, exb, den, e, etot);
  k_fill<<<2048, blk, 0, stream>>>(o1, (long)n * 128, 0.f);
  k_edge_aggr<2, 64><<<gAggr, blk, 0, stream>>>(esrc, edst, hbuf, exb, den, o1, e, etot);
  k_fill<<<1, 32, 0, stream>>>(stats, 2, 0.f);
  k_bias_stats<<<gElem, blk, 0, stream>>>(o1, b1, n * 128, stats);
  k_ln_apply<true><<<gElem, blk, 0, stream>>>(o1, ln1w, ln1b, stats, n * 128);

  // ---------------- layer 2: GATConv(128 -> 128, heads=1) ----------------
  k_gemm128_wmma<<<gTile, blk, 0, stream>>>(o1, W2, hbuf, n);
  k_att<1, 128><<<gN1, blk, 0, stream>>>(hbuf, as2, ad2, asb, adb, n);
  k_fill<<<512, blk, 0, stream>>>(emx, n, -1e30f);
  k_fill<<<512, blk, 0, stream>>>(den, n, 0.f);
  k_edge_max<1><<<gEdge, blk, 0, stream>>>(esrc, edst, asb, adb, emx, e, etot);
  k_edge_exp<1><<<gEdge, blk, 0, stream>>>(esrc, edst, asb, adb, emx, exb, den, e, etot);
  k_fill<<<2048, blk, 0, stream>>>(o2, (long)n * 128, 0.f);
  k_edge_aggr<1, 128><<<gAggr, blk, 0, stream>>>(esrc, edst, hbuf, exb, den, o2, e, etot);
  k_fill<<<1, 32, 0, stream>>>(stats, 2, 0.f);
  k_bias_stats<<<gElem, blk, 0, stream>>>(o2, b2, n * 128, stats);
  k_ln_apply<false><<<gElem, blk, 0, stream>>>(o2, ln2w, ln2b, stats, n * 128);

  // ---------------- SimpleConv(mean) over original edges + residual + relu ------
  k_fill<<<2048, blk, 0, stream>>>(out, (long)n * 128, 0.f);
  k_fill<<<512, blk, 0, stream>>>(cntb, n, 0.f);
  k_cnt<<<gE, blk, 0, stream>>>(edst, cntb, e);
  k_sc_aggr<<<gE4, blk, 0, stream>>>(esrc, edst, o2, out, e);
  k_final<<<gElem, blk, 0, stream>>>(out, cntb, x, n * 128);
}